// ConnectivityLoss_36429912605423
// MI455X (gfx1250) — compile-verified
//
#include <hip/hip_runtime.h>
#include <stdint.h>

typedef __attribute__((ext_vector_type(2))) float v2f;
typedef __attribute__((ext_vector_type(8))) float v8f;
typedef __attribute__((ext_vector_type(4))) unsigned int u32x4;
typedef __attribute__((ext_vector_type(8))) int i32x8;
typedef __attribute__((ext_vector_type(4))) int i32x4;

#define NTHREADS 256
#define NN 1024      // nodes per 32x32 window
#define NE 1984      // edges (992 horizontal + 992 vertical)
#define HALF 992
#define NSORT 2048

// ---- edge decode: e<992 horizontal (i=e/31, j=e%31, a=i*32+j, b=a+1);
//      e>=992 vertical (f=e-992 => a=f, b=f+32)
__device__ __forceinline__ void edge_nodes(int e, int& a, int& b) {
    if (e < HALF) { int i = e / 31; int j = e - i * 31; a = (i << 5) + j; b = a + 1; }
    else          { int f = e - HALF; a = f; b = f + 32; }
}

__device__ __forceinline__ int uf_find(unsigned short* p, int x) {
    while ((int)p[x] != x) { p[x] = p[p[x]]; x = (int)p[x]; }
    return x;
}

// Serial Kruskal maximin pass (runs on ONE lane). Label-overlap maps are
// linked lists in a node-indexed pool (splice-merge, no allocation); dot
// product via direct-address tmp[] publish of the smaller list.
__device__ void kruskal_run(const unsigned long long* keys,
                            unsigned short* parent, unsigned short* tot,
                            short* head, unsigned short* lab, unsigned short* cnt,
                            short* nxt, unsigned short* len, unsigned short* tmp,
                            unsigned int* w, int pos)
{
    for (int t = 0; t < NSORT; ++t) {
        unsigned long long k = keys[t];
        int e = 2047 - (int)(k & 2047ull);
        if (e >= NE) continue;           // padding entries
        int a, b; edge_nodes(e, a, b);
        int ra = uf_find(parent, a), rb = uf_find(parent, b);
        if (ra == rb) continue;
        unsigned int ta = tot[ra], tb = tot[rb];
        int la = (int)len[ra], lb = (int)len[rb];
        int bigHead, smallHead;
        if (la >= lb) { bigHead = (int)head[ra]; smallHead = (int)head[rb]; }
        else          { bigHead = (int)head[rb]; smallHead = (int)head[ra]; }
        // publish small
        for (int p2 = smallHead; p2 >= 0; p2 = (int)nxt[p2]) tmp[lab[p2]] = cnt[p2];
        // absorb into big, compute dot over common labels
        unsigned int dot = 0; int common = 0;
        for (int p2 = bigHead; p2 >= 0; p2 = (int)nxt[p2]) {
            unsigned int t2 = (unsigned int)tmp[lab[p2]];
            if (t2) {
                dot += (unsigned int)cnt[p2] * t2;
                cnt[p2] = (unsigned short)((unsigned int)cnt[p2] + t2);
                tmp[lab[p2]] = 0; ++common;
            }
        }
        // splice leftover (new-label) entries of small into big; clear tmp
        int p2 = smallHead;
        while (p2 >= 0) {
            int nx = (int)nxt[p2];
            if (tmp[lab[p2]]) { tmp[lab[p2]] = 0; nxt[p2] = (short)bigHead; bigHead = p2; }
            p2 = nx;
        }
        w[e] = pos ? dot : (ta * tb - dot);
        parent[rb] = (unsigned short)ra;
        tot[ra] = (unsigned short)(ta + tb);
        head[ra] = (short)bigHead; head[rb] = (short)-1;
        len[ra] = (unsigned short)(la + lb - common); len[rb] = 0;
    }
}

__global__ void __launch_bounds__(NTHREADS)
malis_window_kernel(const float* __restrict__ y_true,
                    const float* __restrict__ y_pred,
                    float* __restrict__ out)
{
    extern __shared__ unsigned char smem[];
    unsigned long long* keysN = (unsigned long long*)smem;
    unsigned long long* keysP = keysN + NSORT;
    float* shP = (float*)(keysP + NSORT);
    float* shT = shP + NN;
    unsigned int* wN = (unsigned int*)(shT + NN);
    unsigned int* wP = wN + NE;
    unsigned short* seg     = (unsigned short*)(wP + NE);
    unsigned short* parentN = seg + NN;
    unsigned short* parentP = parentN + NN;
    unsigned short* totN    = parentP + NN;
    unsigned short* totP    = totN + NN;
    short*          headN   = (short*)(totP + NN);
    short*          headP   = headN + NN;
    unsigned short* labN    = (unsigned short*)(headP + NN);
    unsigned short* labP    = labN + NN;
    unsigned short* cntN    = labP + NN;
    unsigned short* cntP    = cntN + NN;
    short*          nxtN    = (short*)(cntP + NN);
    short*          nxtP    = nxtN + NN;
    unsigned short* lenN    = (unsigned short*)(nxtP + NN);
    unsigned short* lenP    = lenN + NN;
    unsigned short* tmpN    = lenP + NN;            // NN+2 entries
    unsigned short* tmpP    = tmpN + (NN + 2);
    float* red     = (float*)(tmpP + (NN + 2));     // 32 floats (4 groups x 8 waves)
    float* waveSum = red + 32;                       // 8 floats
    int* flag = (int*)(waveSum + 8);
    int* skip = flag + 1;
    unsigned int* sN = (unsigned int*)(skip + 1);
    unsigned int* sP = sN + 1;

    const int tid = threadIdx.x;
    const int blk = blockIdx.x;
    const int bb = blk >> 6;
    const int wr = (blk >> 3) & 7;
    const int wc = blk & 7;
    const size_t baseElem = (size_t)bb * 65536 + (size_t)wr * 32 * 256 + (size_t)wc * 32;

    // ---- Phase 1: TDM async-tensor load of the two 32x32 f32 tiles into LDS
    if (tid < 32) {
        unsigned long long gp = (unsigned long long)(uintptr_t)(y_pred + baseElem);
        unsigned long long gt = (unsigned long long)(uintptr_t)(y_true + baseElem);
        unsigned int ldsP = (unsigned int)(uintptr_t)(void*)shP;
        unsigned int ldsT = (unsigned int)(uintptr_t)(void*)shT;
        // D# group0: count=1, lds_addr, global_addr[56:0], type=2 (bits 127:126)
        u32x4 g0p = { 1u, ldsP, (unsigned int)gp,
                      ((unsigned int)(gp >> 32) & 0x01FFFFFFu) | 0x80000000u };
        u32x4 g0t = { 1u, ldsT, (unsigned int)gt,
                      ((unsigned int)(gt >> 32) & 0x01FFFFFFu) | 0x80000000u };
        // D# group1: data_size=4B; tensor 256x256; tile 32x32; dim0 stride 256
        i32x8 g1 = { 0x00020000,           // data_size=2 (4 bytes)
                     (int)(256u << 16),    // tensor_dim0 low16 in [63:48]
                     (int)(256u << 16),    // tensor_dim1 low16 in [95:80]
                     (int)(32u << 16),     // tile_dim0 in [127:112]
                     32,                   // tile_dim1 in [143:128]
                     256,                  // tensor_dim0_stride low32
                     0, 0 };
        i32x4 z4 = { 0, 0, 0, 0 };
        i32x8 z8 = { 0, 0, 0, 0, 0, 0, 0, 0 };
        __builtin_amdgcn_tensor_load_to_lds(g0p, g1, z4, z4, z8, 0);
        __builtin_amdgcn_tensor_load_to_lds(g0t, g1, z4, z4, z8, 0);
        __builtin_amdgcn_s_wait_tensorcnt(0);
    }
    __syncthreads();

    // ---- Phase 2: window skip checks (exact float compares, like reference)
    {
        float pmn = 3.402823466e38f, pmx = -3.402823466e38f;
        float tmn = 3.402823466e38f, tmx = -3.402823466e38f;
        for (int v = tid; v < NN; v += NTHREADS) {
            float p = shP[v], t = shT[v];
            pmn = fminf(pmn, p); pmx = fmaxf(pmx, p);
            tmn = fminf(tmn, t); tmx = fmaxf(tmx, t);
        }
        for (int o = 16; o; o >>= 1) {
            pmn = fminf(pmn, __shfl_xor(pmn, o, 32));
            pmx = fmaxf(pmx, __shfl_xor(pmx, o, 32));
            tmn = fminf(tmn, __shfl_xor(tmn, o, 32));
            tmx = fmaxf(tmx, __shfl_xor(tmx, o, 32));
        }
        int wid = tid >> 5;
        if ((tid & 31) == 0) { red[wid] = pmn; red[8+wid] = pmx; red[16+wid] = tmn; red[24+wid] = tmx; }
        __syncthreads();
        if (tid == 0) {
            float a0 = red[0], a1 = red[8], a2 = red[16], a3 = red[24];
            for (int i = 1; i < 8; ++i) {
                a0 = fminf(a0, red[i]); a1 = fmaxf(a1, red[8+i]);
                a2 = fminf(a2, red[16+i]); a3 = fmaxf(a3, red[24+i]);
            }
            *skip = (a0 == 1.0f || a1 == 0.0f || a2 == 1.0f || a3 == 0.0f) ? 1 : 0;
        }
        __syncthreads();
        if (*skip) return;
    }

    // ---- Phase 3: 8-connected CCL of (T==0) via min-label propagation
    for (int v = tid; v < NN; v += NTHREADS)
        seg[v] = (shT[v] == 0.0f) ? (unsigned short)(v + 1) : (unsigned short)0;
    __syncthreads();
    for (;;) {
        if (tid == 0) *flag = 0;
        __syncthreads();
        bool any = false;
        for (int v = tid; v < NN; v += NTHREADS) {
            unsigned short l = seg[v];
            if (!l) continue;
            int r = v >> 5, c = v & 31;
            unsigned short m = l;
            for (int dr = -1; dr <= 1; ++dr)
                for (int dc = -1; dc <= 1; ++dc) {
                    if (dr == 0 && dc == 0) continue;
                    int rr = r + dr, cc = c + dc;
                    if ((unsigned)rr > 31u || (unsigned)cc > 31u) continue;
                    unsigned short ln = seg[(rr << 5) + cc];
                    if (ln && ln < m) m = ln;
                }
            if (m < l) { seg[v] = m; any = true; }
        }
        if (any) *flag = 1;
        __syncthreads();
        int f = *flag;
        __syncthreads();
        if (!f) break;
    }

    // ---- Phase 4: build sort keys (stable descending argsort encoding) + init UF
    for (int e = tid; e < NE; e += NTHREADS) {
        int a, b; edge_nodes(e, a, b);
        float ca = shP[a] + shP[b];
        float g  = shT[a] + shT[b];
        float cn = (g > 20.0f) ? 20.0f : ca;   // costs_n clamp
        float cp = (g < 10.0f) ? 0.0f  : ca;   // costs_p zero
        unsigned long long idx = (unsigned long long)(2047 - e);
        keysN[e] = ((unsigned long long)__float_as_uint(cn) << 11) | idx;
        keysP[e] = ((unsigned long long)__float_as_uint(cp) << 11) | idx;
        wN[e] = 0u; wP[e] = 0u;
    }
    for (int e = NE + tid; e < NSORT; e += NTHREADS) { keysN[e] = 0ull; keysP[e] = 0ull; }
    for (int v = tid; v < NN; v += NTHREADS) {
        parentN[v] = (unsigned short)v; parentP[v] = (unsigned short)v;
        unsigned short s = seg[v];
        labN[v] = s; labP[v] = s;
        cntN[v] = 1; cntP[v] = 1;
        nxtN[v] = -1; nxtP[v] = -1;
        headN[v] = s ? (short)v : (short)-1;
        headP[v] = s ? (short)v : (short)-1;
        totN[v] = s ? 1 : 0; totP[v] = s ? 1 : 0;
        lenN[v] = s ? 1 : 0; lenP[v] = s ? 1 : 0;
        tmpN[v] = 0; tmpP[v] = 0;
    }
    if (tid == 0) { tmpN[NN] = 0; tmpN[NN+1] = 0; tmpP[NN] = 0; tmpP[NN+1] = 0; }
    __syncthreads();

    // ---- Phase 5: bitonic sort (descending) both key arrays, 2048 u64 each
    for (int k = 2; k <= NSORT; k <<= 1) {
        for (int j = k >> 1; j > 0; j >>= 1) {
            __syncthreads();
            for (int i = tid; i < NSORT; i += NTHREADS) {
                int ixj = i ^ j;
                if (ixj > i) {
                    bool up = ((i & k) == 0);
                    unsigned long long x = keysN[i], y = keysN[ixj];
                    if (up ? (x < y) : (x > y)) { keysN[i] = y; keysN[ixj] = x; }
                    x = keysP[i]; y = keysP[ixj];
                    if (up ? (x < y) : (x > y)) { keysP[i] = y; keysP[ixj] = x; }
                }
            }
        }
    }
    __syncthreads();

    // ---- Phase 6: two independent serial Kruskal passes on separate waves
    if (tid == 0)
        kruskal_run(keysN, parentN, totN, headN, labN, cntN, nxtN, lenN, tmpN, wN, 0);
    if (tid == 128)
        kruskal_run(keysP, parentP, totP, headP, labP, cntP, nxtP, lenP, tmpP, wP, 1);
    __syncthreads();

    // ---- Phase 7: weight sums, per-edge normalized+masked loss contribution
    if (tid == 0) { *sN = 0u; *sP = 0u; }
    __syncthreads();
    {
        unsigned int accN = 0, accP = 0;
        for (int e = tid; e < NE; e += NTHREADS) { accN += wN[e]; accP += wP[e]; }
        atomicAdd(sN, accN); atomicAdd(sP, accP);
    }
    __syncthreads();
    double dsn = (double)(*sN), dsp = (double)(*sP);
    double part = 0.0;
    for (int e = tid; e < NE; e += NTHREADS) {
        int a, b; edge_nodes(e, a, b);
        float g  = shT[a] + shT[b];
        float gc = fminf(g, 20.0f);
        double ewn = 0.0, ewp = 0.0;
        if (dsn > 0.0 && gc < 10.0f)  ewn = (double)wN[e] / dsn;  // ewn[gt>=10]=0
        if (dsp > 0.0 && gc >= 20.0f) ewp = (double)wP[e] / dsp;  // ewp[gt<20]=0
        float wnf = (float)ewn, wpf = (float)ewp;
        float pa = shP[a], pb = shP[b];
        float qa = 20.0f - pa, qb = 20.0f - pb;
        part += (double)(wnf * (pa * pa + pb * pb)) + (double)(wpf * (qa * qa + qb * qb));
    }
    for (int o = 16; o; o >>= 1) part += __shfl_down(part, o, 32);
    if ((tid & 31) == 0) waveSum[tid >> 5] = (float)part;
    __syncthreads();

    // ---- Phase 8: final 8-value reduction via WMMA ones trick (wave 0 only)
    if (tid < 32) {
        float a0 = (tid < 8) ? waveSum[tid] : 0.0f;
        v2f A = { a0, 0.0f };      // A[m][k]: rows 0..7 carry wave sums (k=0)
        v2f Bv = { 1.0f, 1.0f };   // B = all ones -> D[m][n] = row-sum(A)
        v8f C = { 0.f, 0.f, 0.f, 0.f, 0.f, 0.f, 0.f, 0.f };
        C = __builtin_amdgcn_wmma_f32_16x16x4_f32(false, A, false, Bv,
                                                  (short)0, C, false, false);
        if (tid == 0) {
            float tot = C[0] + C[1] + C[2] + C[3] + C[4] + C[5] + C[6] + C[7];
            atomicAdd(out, tot);
        }
    }
}

__global__ void zero_out_kernel(float* out) {
    if (threadIdx.x == 0) out[0] = 0.0f;
}

extern "C" void kernel_launch(void* const* d_in, const int* in_sizes, int n_in,
                              void* d_out, int out_size, void* d_ws, size_t ws_size,
                              hipStream_t stream) {
    (void)n_in; (void)out_size; (void)d_ws; (void)ws_size;
    const float* y_true = (const float*)d_in[0];
    const float* y_pred = (const float*)d_in[1];
    float* out = (float*)d_out;

    int B = in_sizes[0] / (256 * 256);
    if (B < 1) B = 1;
    int nblk = B * 64;   // 8x8 windows of 32x32 per batch item

    // LDS bytes: 2*16384 (keys) + 2*4096 (P,T) + 2*7936 (wN,wP)
    //          + 2048 (seg) + 14*2048 (UF arrays) + 2*2052 (tmp) + 176 (misc)
    constexpr size_t SMEM_BYTES = 94208;   // 92 KB (>= 91832 actually used)

    zero_out_kernel<<<1, 32, 0, stream>>>(out);
    malis_window_kernel<<<nblk, NTHREADS, SMEM_BYTES, stream>>>(y_true, y_pred, out);
}